// MyMCBLayer_52510270161274
// MI455X (gfx1250) — compile-verified
//
#include <hip/hip_runtime.h>
#include <math.h>

// ---------------------------------------------------------------------------
// MCB (compact bilinear pooling) fused per-row kernel for MI455X (gfx1250)
//   sk = count-sketch(v1) + i*count-sketch(v2)   (LDS ds_add_f32 scatter)
//   F  = FFT_8192(sk)        [4-step: WMMA DFT-16 stage + 16x Stockham-512]
//   P  = F1*F2 via conjugate-symmetry split (saves a whole FFT)
//   z  = IFFT(P) = conj(FFT(conj(P)))/N
//   y  = Re(z)*|z|^{-1/2}; global L2 normalize in a second (L2-resident) pass
// ---------------------------------------------------------------------------

typedef __attribute__((ext_vector_type(2))) float v2f;
typedef __attribute__((ext_vector_type(8))) float v8f;

#define D_FFT 8192
#define TPB   512
#define PI_F  3.14159265358979323846f

// F[k] lives at row (k&15), column (k>>4) after the 4-step FFT
__device__ __forceinline__ int idxF(int k) { return ((k & 15) << 9) | (k >> 4); }

// ---------------------------------------------------------------------------
// Stage A: Y[k1][n2] = sum_{n1} w16^{k1*n1} * x[n1*512 + n2]   (complex)
// Dense 16x16 DFT matrix applied to a 16x512 operand via v_wmma_f32_16x16x4.
// Each wave owns 16-column tiles; 4 K-chunks x 4 real matmuls per tile.
// ---------------------------------------------------------------------------
__device__ __forceinline__ void dft16_stage(const float* __restrict__ sr,
                                            const float* __restrict__ si,
                                            float* __restrict__ dr,
                                            float* __restrict__ di,
                                            int tid)
{
  const int lane = tid & 31;
  const int wave = tid >> 5;
  const int r16  = lane & 15;   // A: row k1 ; B/D: column within tile
  const int hi   = lane >> 4;   // which K half (A/B) / row half (D)

  for (int t = wave; t < 32; t += 16) {      // 32 column tiles, 16 waves
    const int c0  = t << 4;
    const int col = c0 + r16;
    v8f accR = {0.f,0.f,0.f,0.f,0.f,0.f,0.f,0.f};
    v8f accI = {0.f,0.f,0.f,0.f,0.f,0.f,0.f,0.f};
#pragma unroll
    for (int kk = 0; kk < 4; ++kk) {
      const int kb = (kk << 2) + (hi << 1);  // K index held in .x (.y = kb+1)
      float s0, c0w, s1, c1w;
      __sincosf((2.0f*PI_F/16.0f) * (float)(r16 * kb),     &s0, &c0w);
      __sincosf((2.0f*PI_F/16.0f) * (float)(r16 * (kb+1)), &s1, &c1w);
      // DFT16 = cos(th) - i*sin(th). f32 WMMA has no A-neg modifier
      // (NEG is C-only per ISA), so carry signs in the generated values.
      v2f Ar, Ain, Aip;
      Ar.x  =  c0w; Ar.y  =  c1w;   // Re(D16)
      Ain.x = -s0;  Ain.y = -s1;    // Im(D16) = -sin
      Aip.x =  s0;  Aip.y =  s1;    // -Im(D16) = +sin
      v2f Br, Bi;
      Br.x = sr[(kb<<9) + col];  Br.y = sr[((kb+1)<<9) + col];
      Bi.x = si[(kb<<9) + col];  Bi.y = si[((kb+1)<<9) + col];
      // Yr = Dr*Mr - Di*Mi ; Yi = Dr*Mi + Di*Mr
      accR = __builtin_amdgcn_wmma_f32_16x16x4_f32(false, Ar,  false, Br, (short)0, accR, false, false);
      accR = __builtin_amdgcn_wmma_f32_16x16x4_f32(false, Aip, false, Bi, (short)0, accR, false, false);
      accI = __builtin_amdgcn_wmma_f32_16x16x4_f32(false, Ar,  false, Bi, (short)0, accI, false, false);
      accI = __builtin_amdgcn_wmma_f32_16x16x4_f32(false, Ain, false, Br, (short)0, accI, false, false);
    }
#pragma unroll
    for (int v = 0; v < 8; ++v) {            // D: VGPR v -> rows v, v+8
      const int orow = v + (hi << 3);
      dr[(orow<<9) + col] = accR[v];
      di[(orow<<9) + col] = accI[v];
    }
  }
}

__device__ __forceinline__ void twiddle_stage(float* yr, float* yi, int tid)
{
  for (int idx = tid; idx < D_FFT; idx += TPB) {
    const int k1 = idx >> 9, n2 = idx & 511;
    float s, c;
    __sincosf((-2.0f*PI_F/(float)D_FFT) * (float)(k1 * n2), &s, &c);
    const float a = yr[idx], b = yi[idx];
    yr[idx] = a*c - b*s;
    yi[idx] = a*s + b*c;
  }
}

// ---------------------------------------------------------------------------
// Full 8192-point FFT. Input natural order in (xr,xi); (yr,yi) is scratch.
// Result lands back in (xr,xi) at layout idxF(k) = (k&15)*512 + (k>>4).
// Step C: 16 independent 512-pt Stockham FFTs, one wave per row (self-sorting,
// ping-pong between the two LDS buffers; 9 radix-2 stages -> ends in x).
// ---------------------------------------------------------------------------
__device__ __forceinline__ void fft8192(float* xr, float* xi,
                                        float* yr, float* yi, int tid)
{
  dft16_stage(xr, xi, yr, yi, tid);
  __syncthreads();
  twiddle_stage(yr, yi, tid);
  __syncthreads();

  const int wv = tid >> 5, ln = tid & 31;
  const int base = wv << 9;
  int ncur = 512, s = 1, l2s = 0;
  for (int st = 0; st < 9; ++st) {
    float* sr  = (st & 1) ? xr : yr;
    float* si_ = (st & 1) ? xi : yi;
    float* dr  = (st & 1) ? yr : xr;
    float* di_ = (st & 1) ? yi : xi;
    const int m = ncur >> 1;
    const float th = -2.0f*PI_F/(float)ncur;
    for (int t = ln; t < 256; t += 32) {     // 8 butterflies per lane
      const int q = t & (s - 1);
      const int p = t >> l2s;
      float ws, wc; __sincosf(th * (float)p, &ws, &wc);
      const int ia = base + q + (p << l2s);
      const int ib = ia + (m << l2s);
      const float ar = sr[ia], ai = si_[ia];
      const float br = sr[ib], bi = si_[ib];
      const int oa = base + q + (p << (l2s + 1));
      const int ob = oa + s;
      dr[oa] = ar + br;  di_[oa] = ai + bi;
      const float ur = ar - br, ui = ai - bi;
      dr[ob] = ur*wc - ui*ws;
      di_[ob] = ur*ws + ui*wc;
    }
    __syncthreads();
    ncur >>= 1; s <<= 1; ++l2s;
  }
}

extern "C" __global__ void __launch_bounds__(TPB)
mcb_row_kernel(const float* __restrict__ v1, const float* __restrict__ v2,
               const float* __restrict__ s1, const float* __restrict__ s2,
               const int* __restrict__ h1, const int* __restrict__ h2,
               float* __restrict__ out, float* __restrict__ norm_acc,
               int n1, int n2)
{
  extern __shared__ float smem[];            // 4 x 8192 floats = 128 KB
  float* Ar = smem;
  float* Ai = smem +     D_FFT;
  float* Br = smem + 2 * D_FFT;
  float* Bi = smem + 3 * D_FFT;
  const int tid = threadIdx.x;
  const int b   = blockIdx.x;

  for (int i = tid; i < D_FFT; i += TPB) { Ar[i] = 0.f; Ai[i] = 0.f; }
  __syncthreads();

  // count-sketch scatter: sk1 -> Re, sk2 -> Im (two real FFTs in one complex)
  const float* v1r = v1 + (size_t)b * n1;
  const float* v2r = v2 + (size_t)b * n2;
  for (int j = tid; j < n1; j += TPB) atomicAdd(&Ar[h1[j]], s1[j] * v1r[j]);
  for (int j = tid; j < n2; j += TPB) atomicAdd(&Ai[h2[j]], s2[j] * v2r[j]);
  __syncthreads();

  fft8192(Ar, Ai, Br, Bi, tid);              // F in (Ar,Ai) @ idxF layout

  // split F1/F2 by conjugate symmetry, multiply, store conj(P) natural -> B
  for (int k = tid; k < D_FFT; k += TPB) {
    const int nk = (D_FFT - k) & (D_FFT - 1);
    const int ik = idxF(k), ink = idxF(nk);
    const float ar_ = Ar[ik],  ai_ = Ai[ik];
    const float br_ = Ar[ink], bi_ = Ai[ink];
    const float f1r = 0.5f*(ar_ + br_), f1i = 0.5f*(ai_ - bi_);
    const float dre = ar_ - br_,        dim = ai_ + bi_;
    const float f2r = 0.5f*dim,         f2i = -0.5f*dre;   // (d)*( -i/2 )
    const float pr  = f1r*f2r - f1i*f2i;
    const float pi_ = f1r*f2i + f1i*f2r;
    Br[k] = pr; Bi[k] = -pi_;                // conj for IFFT-via-FFT
  }
  __syncthreads();

  fft8192(Br, Bi, Ar, Ai, tid);              // G in (Br,Bi) @ idxF layout

  // z = conj(G)/N ; y = Re(z)*|z|^{-1/2} ; write + local sum of squares
  const float invN = 1.0f / (float)D_FFT;
  float ssq = 0.f;
  float* orow = out + (size_t)b * D_FFT;
  for (int k = tid; k < D_FFT; k += TPB) {
    const int ik = idxF(k);
    const float zr = Br[ik] * invN;
    const float zi = -Bi[ik] * invN;
    const float mag = sqrtf(zr*zr + zi*zi);
    const float val = (mag > 0.f) ? zr * rsqrtf(mag) : 0.f;
    orow[k] = val;
    ssq += val * val;
  }
  // block reduction -> one global fp32 atomic per row
#pragma unroll
  for (int off = 16; off > 0; off >>= 1) ssq += __shfl_down(ssq, off, 32);
  if ((tid & 31) == 0) Ar[tid >> 5] = ssq;   // Ar is dead scratch here
  __syncthreads();
  if (tid < 32) {
    float s = (tid < 16) ? Ar[tid] : 0.f;
#pragma unroll
    for (int off = 8; off > 0; off >>= 1) s += __shfl_down(s, off, 32);
    if (tid == 0) atomicAdd(norm_acc, s);
  }
}

extern "C" __global__ void zero_acc_kernel(float* acc)
{
  if (threadIdx.x == 0 && blockIdx.x == 0) *acc = 0.f;
}

// second pass: 128 MB output fits in the 192 MB L2, so this is L2-resident
extern "C" __global__ void __launch_bounds__(256)
norm_scale_kernel(float* __restrict__ out, const float* __restrict__ acc, int n4)
{
  const float scale = rsqrtf(fmaxf(*acc, 1e-12f));
  const int i = blockIdx.x * 256 + threadIdx.x;
  if (i < n4) {
    float4* o4 = (float4*)out;
    float4 v = o4[i];
    v.x *= scale; v.y *= scale; v.z *= scale; v.w *= scale;
    o4[i] = v;
  }
}

extern "C" void kernel_launch(void* const* d_in, const int* in_sizes, int n_in,
                              void* d_out, int out_size, void* d_ws, size_t ws_size,
                              hipStream_t stream)
{
  const float* v1 = (const float*)d_in[0];
  const float* v2 = (const float*)d_in[1];
  const float* s1 = (const float*)d_in[2];
  const float* s2 = (const float*)d_in[3];
  const int*   h1 = (const int*)d_in[4];
  const int*   h2 = (const int*)d_in[5];
  float* out = (float*)d_out;
  float* acc = (float*)d_ws;                 // 1 float accumulator

  const int n1 = in_sizes[2];                // 2048
  const int n2 = in_sizes[3];                // 2048
  const int B  = in_sizes[0] / n1;           // 4096

  zero_acc_kernel<<<1, 64, 0, stream>>>(acc);
  mcb_row_kernel<<<B, TPB, 4 * D_FFT * sizeof(float), stream>>>(
      v1, v2, s1, s2, h1, h2, out, acc, n1, n2);
  const int n4 = out_size / 4;
  norm_scale_kernel<<<(n4 + 255) / 256, 256, 0, stream>>>(out, acc, n4);
}